// Device_73349451481210
// MI455X (gfx1250) — compile-verified
//
#include <hip/hip_runtime.h>
#include <hip/hip_bf16.h>
#include <stdint.h>

typedef __attribute__((ext_vector_type(2))) float v2f;
typedef __attribute__((ext_vector_type(8))) float v8f;

#define PKT 256
#define INIT_OVH 1e-4f

// ---------------------------------------------------------------------------
// Kernel A: per-256-element chunk inclusive prefix sums of net = harvest - L.
// One wave per chunk. Data staged into LDS via gfx1250 async global->LDS
// copies (ASYNCcnt), then the 16x16 chunk matrix X is multiplied by an
// upper-triangular ones matrix U using 4 chained V_WMMA_F32_16X16X4_F32
// (K-slices of 4) -> P[r,c] = prefix within row r. Row offsets are combined
// through LDS to produce the full 256-wide inclusive prefix.
// ---------------------------------------------------------------------------
__global__ __launch_bounds__(32) void chunk_prefix_wmma(
    const float* __restrict__ eh, const float* __restrict__ leak,
    float* __restrict__ pref)
{
  __shared__ float s[256];
  __shared__ float rowsum[16];
  const int lane  = threadIdx.x;
  const int chunk = blockIdx.x;
  const float L = leak[0];

  // ---- async copy: 8 x (32 lanes x b32) = 256 floats into LDS -------------
  {
    const float* g = eh + (size_t)chunk * 256 + lane;
    // low 32 bits of the flat shared-aperture address == LDS byte offset
    unsigned lds_off = (unsigned)(uintptr_t)(&s[lane]);
    asm volatile(
        "global_load_async_to_lds_b32 %0, %1, off\n\t"
        "global_load_async_to_lds_b32 %0, %1, off offset:128\n\t"
        "global_load_async_to_lds_b32 %0, %1, off offset:256\n\t"
        "global_load_async_to_lds_b32 %0, %1, off offset:384\n\t"
        "global_load_async_to_lds_b32 %0, %1, off offset:512\n\t"
        "global_load_async_to_lds_b32 %0, %1, off offset:640\n\t"
        "global_load_async_to_lds_b32 %0, %1, off offset:768\n\t"
        "global_load_async_to_lds_b32 %0, %1, off offset:896\n\t"
        "s_wait_asynccnt 0x0"
        :
        : "v"(lds_off), "v"(g)
        : "memory");
  }
  __syncthreads();

  const int hgrp = lane >> 4;   // 0: lanes 0-15, 1: lanes 16-31
  const int lid  = lane & 15;

  // ---- P = X * U  via 4 accumulating WMMAs (K = 4 each) -------------------
  // A 16x4 f32 layout: lanes 0-15 rows M, VGPR0/1 = K+0/K+1; lanes 16-31 K+2/K+3.
  // B 4x16 f32 layout: row-striped across lanes analogously.
  v8f acc = {0.f,0.f,0.f,0.f,0.f,0.f,0.f,0.f};
  for (int t = 0; t < 4; ++t) {
    const int ka = 4 * t + 2 * hgrp;     // K index base held by this lane half
    v2f a, b;
    a.x = s[16 * lid + ka + 0] - L;      // X[row=lid, ka+0]
    a.y = s[16 * lid + ka + 1] - L;      // X[row=lid, ka+1]
    b.x = ((ka + 0) <= lid) ? 1.0f : 0.0f;  // U[ka+0, n=lid]
    b.y = ((ka + 1) <= lid) ? 1.0f : 0.0f;  // U[ka+1, n=lid]
    acc = __builtin_amdgcn_wmma_f32_16x16x4_f32(
        /*neg_a=*/false, a, /*neg_b=*/false, b,
        /*c_mod=*/(short)0, acc, /*reuse_a=*/false, /*reuse_b=*/false);
  }

  // ---- combine row offsets: S[r] = P[r,15] lives in lanes 15 / 31 ---------
  if (lane == 15) {
    for (int v = 0; v < 8; ++v) rowsum[v] = acc[v];
  }
  if (lane == 31) {
    for (int v = 0; v < 8; ++v) rowsum[8 + v] = acc[v];
  }
  __syncthreads();

  float offv[8];
  {
    float run = 0.f;
    for (int r = 0; r < 16; ++r) {
      if (r >= 8 * hgrp && r < 8 * hgrp + 8) offv[r - 8 * hgrp] = run;
      run += rowsum[r];
    }
  }

  float* outp = pref + (size_t)chunk * 256;
  for (int v = 0; v < 8; ++v) {
    const int r = 8 * hgrp + v;          // C/D layout: VGPR v -> row v or v+8
    outp[16 * r + lid] = acc[v] + offv[v];
  }
}

// ---------------------------------------------------------------------------
// Kernel B: sequential controller (one wave). Faithful scalar state machine,
// with three vectorized regimes:
//   * packet windows (256 steps): closed-form, 32-wide fill, exact
//   * halted tail: bulk fill
//   * speculative 32-wide advance in states 1/2 using the WMMA chunk prefix:
//     lanes compute candidate energies, __ballot validates that no event
//     (tx threshold / clip-hi / clip-lo / state-1 exit) fires in the window.
// ---------------------------------------------------------------------------
__global__ __launch_bounds__(32) void ehd_scan(
    const float* __restrict__ eh, const float* __restrict__ leak,
    const float* __restrict__ thr, const float* __restrict__ alp,
    const float* __restrict__ pref, int use_fast,
    float* __restrict__ e_out, float* __restrict__ v_out,
    float* __restrict__ a_out, int T)
{
  const int lane = threadIdx.x;
  const float L  = leak[0];
  const float th = thr[0];
  const float a  = alp[0];
  const float max_e = 4.0f * th;
  const float on_th = 5.0f * L + INIT_OVH;
  const float tx_th = th + a + 5.0f * L;
  const float lin_scale = (th + a) * (1.0f / 256.0f);  // lin[i] = scale*(i+1)
  const float QNAN = __builtin_nanf("");

  if (lane == 0) { e_out[0] = 0.f; v_out[0] = QNAN; a_out[0] = 0.f; }

  int   state = 0, pkt = 0;
  bool  halted = false, skipf = false;
  float e_pkt = 0.f, e_prev = 0.f;
  int   k = 1;

  while (k < T) {
    if (halted) {                                   // bulk-fill the tail
      for (int j = k + lane; j < T; j += 32) {
        e_out[j] = 0.f; v_out[j] = QNAN; a_out[j] = 0.f;
      }
      break;
    }
    if (pkt > 0) {                                  // packet window (pkt==256)
      for (int j = lane; j < pkt; j += 32) {
        float ev = e_pkt - lin_scale * (float)(j + 1) + eh[k + j];
        e_out[k + j] = ev; v_out[k + j] = 1.0f; a_out[k + j] = 0.f;
      }
      e_prev = e_pkt - lin_scale * 256.0f + eh[k + pkt - 1];
      k += pkt; pkt = 0;
      continue;
    }
    if (skipf) {                                    // turn-on overwrite step
      float ev = e_prev - INIT_OVH;
      if (lane == 0) { e_out[k] = ev; v_out[k] = QNAN; a_out[k] = 0.f; }
      e_prev = ev; skipf = false; ++k;
      continue;
    }
    // ---- speculative 32-wide advance ------------------------------------
    if (use_fast && (state == 1 || state == 2) &&
        (k + 32) <= T && ((k & 255) + 32) <= 256) {
      const int chunk_base = k & ~255;
      const int off = k & 255;
      float pb = (off == 0) ? 0.f : pref[chunk_base + off - 1];
      float cand = e_prev + (pref[chunk_base + off + lane] - pb);
      bool bad;
      if (state == 2)
        bad = (cand >= tx_th) | (cand >= max_e) | (cand <= 0.f);
      else  // state 1 stays state 1 only while e >= th+a (and 0 < e < max_e)
        bad = (cand < th + a) | (cand >= max_e) | (cand <= 0.f);
      unsigned long long m = __ballot(bad);
      if ((m & 0xffffffffull) == 0ull) {
        e_out[k + lane] = cand; v_out[k + lane] = QNAN; a_out[k + lane] = 0.f;
        e_prev = __shfl(cand, 31, 32);
        __builtin_prefetch(&pref[chunk_base + off + 256], 0, 1);
        k += 32;
        continue;
      }
    }
    // ---- faithful scalar step (uniform across all lanes) -----------------
    float ek = e_prev + eh[k] - L;
    ek = fminf(fmaxf(ek, 0.0f), max_e);
    const bool is_off  = (state == 0);
    const bool is_can  = (state == 1);
    const bool is_cant = (state == 2);
    const bool turn_on  = is_off  && (ek >= on_th);
    const bool tx_cond  = is_cant && (ek >= tx_th);
    const bool tx_bnd   = tx_cond && ((k + PKT + 1) >= T);
    const bool tx_fire  = tx_cond && !tx_bnd;
    const bool can_off  = is_can  && (ek == 0.f);
    const bool can_cant = is_can  && (ek > 0.f) && (ek < th + a);
    const bool cant_off = is_cant && !tx_cond && (ek == 0.f);
    int ns = state;
    if (turn_on || can_cant) ns = 2;
    if (tx_fire)             ns = 1;
    if (can_off || cant_off) ns = 0;
    state = ns;
    if (tx_fire) { pkt = PKT; e_pkt = ek; }
    halted = halted || tx_bnd;
    skipf = turn_on;
    if (lane == 0) {
      e_out[k] = ek; v_out[k] = QNAN; a_out[k] = tx_fire ? 1.0f : 0.f;
    }
    e_prev = ek; ++k;
  }
}

extern "C" void kernel_launch(void* const* d_in, const int* in_sizes, int n_in,
                              void* d_out, int out_size, void* d_ws, size_t ws_size,
                              hipStream_t stream) {
  const float* eh   = (const float*)d_in[0];
  const float* leak = (const float*)d_in[1];
  const float* thr  = (const float*)d_in[2];
  const float* alp  = (const float*)d_in[3];
  const int T = in_sizes[0];

  float* e_out = (float*)d_out;          // outputs concatenated: e, valid, act
  float* v_out = e_out + T;
  float* a_out = e_out + 2 * T;

  float* pref = (float*)d_ws;
  const int use_fast =
      ((T % 256) == 0 && ws_size >= (size_t)T * sizeof(float)) ? 1 : 0;

  if (use_fast) {
    chunk_prefix_wmma<<<T / 256, 32, 0, stream>>>(eh, leak, pref);
  }
  ehd_scan<<<1, 32, 0, stream>>>(eh, leak, thr, alp, pref, use_fast,
                                 e_out, v_out, a_out, T);
}